// FeatureEmbeddingBank_77498389889625
// MI455X (gfx1250) — compile-verified
//
#include <hip/hip_runtime.h>
#include <hip/hip_bf16.h>

// Embedding-bag gather + mean pool for MI455X (gfx1250, wave32).
//
// Mapping: one bag per HALF-WAVE. 16 lanes x v4f = 256B = one full D=64
// embedding row, so each global_load_b128 instruction fetches one row for each
// of the wave's two bags (two coalesced 256B requests).
//
// Memory-bound (0.23 flop/byte): the design goals are (a) widest coalesced
// gathers with all 20 row loads in flight, (b) a minimal instruction stream so
// issue bandwidth never competes with the 23.3 TB/s memory system.
//   - index transforms done ONCE per held index (2/lane), rows broadcast via
//     wave32 ds_bpermute (__shfl)
//   - fully branchless (v_cndmask selects, no exec-mask churn in the loop)
//   - bag count via one ballot+popcount per index group
//   - non-temporal store of the write-once output / load of the read-once
//     index stream, preserving L2 for embedding-row reuse.

#define F_REG    25
#define F_TOT    50
#define LL       20
#define DD       64
#define REG_ROWS 50001   // V_REG + 1
#define CMP_ROWS 100001  // N_BUCKETS + 1
#define N_BUCKETS 100000

typedef float v4f __attribute__((ext_vector_type(4)));

__global__ __launch_bounds__(256) void emb_bag_kernel(
    const int*   __restrict__ feats,   // [B, 50*20]
    const float* __restrict__ Wreg,    // [25, 50001, 64]
    const float* __restrict__ Wcmp,    // [25, 100001, 64]
    float*       __restrict__ out,     // [B, 50, 64]
    int nBags)                         // B * 50
{
    const int lane   = threadIdx.x & 31;
    const int sub    = lane & 15;                 // d-chunk within half-wave
    const int waveId = blockIdx.x * (blockDim.x >> 5) + (threadIdx.x >> 5);
    const int bag    = waveId * 2 + (lane >> 4);  // one bag per half-wave
    if (bag >= nBags) return;                     // uniform per half-wave

    const int  b = bag / F_TOT;
    const int  f = bag - b * F_TOT;
    const long ibase = (long)b * (F_TOT * LL) + (long)f * LL;

    // Coalesced, read-once index fetch (non-temporal):
    //   sub 0..15 hold slot 0..15; sub 0..3 hold slot 16..19 (dups elsewhere,
    //   never consumed -> no branch needed).
    const int idxA = __builtin_nontemporal_load(feats + ibase + sub);
    const int idxB = __builtin_nontemporal_load(feats + ibase + 16 + (sub & 3));

    const bool isReg = (f < F_REG);

    // Transform the (at most) 2 indices this lane owns — branchless.
    int rowA, rowB;
    {
        int v    = (idxA > 0) ? idxA : 0;                       // clamp_min(0)
        int rR   = (v < 50000) ? v : 50000;                     // clip to V_REG
        int rC   = (v > 0) ? ((v - 1) % N_BUCKETS) + 1 : 0;     // hash-compress
        rowA     = isReg ? rR : rC;
        v        = (idxB > 0) ? idxB : 0;
        rR       = (v < 50000) ? v : 50000;
        rC       = (v > 0) ? ((v - 1) % N_BUCKETS) + 1 : 0;
        rowB     = isReg ? rR : rC;
    }

    // Bag nonzero-count via wave ballot: bits 0..15 = half 0, 16..31 = half 1.
    const unsigned mA    = (unsigned)__ballot(idxA > 0);
    const unsigned mB    = (unsigned)__ballot((sub < 4) && (idxB > 0));
    const unsigned hmask = (lane & 16) ? 0xFFFF0000u : 0x0000FFFFu;
    int cnt = __popc(mA & hmask) + __popc(mB & hmask);

    // Per-lane gather base: table slice for this feature + this lane's d-chunk.
    const float* tabBase =
        (isReg ? (Wreg + (size_t)f * (REG_ROWS * DD))
               : (Wcmp + (size_t)(f - F_REG) * (CMP_ROWS * DD)))
        + (size_t)sub * 4;

    // Issue all 20 row gathers (global_load_b128) before consuming any.
    // Loop body: 1 ds_bpermute + 1 i64 mad + 1 b128 load.
    v4f vals[LL];
#pragma unroll
    for (int l = 0; l < LL; ++l) {
        const int src = (lane & 16) | (l & 15);   // source lane in SAME half
        const int row = (l < 16) ? __shfl(rowA, src, 32)
                                 : __shfl(rowB, src, 32);
        vals[l] = *reinterpret_cast<const v4f*>(tabBase + (size_t)row * DD);
    }

    v4f acc = (v4f){0.f, 0.f, 0.f, 0.f};
#pragma unroll
    for (int l = 0; l < LL; ++l) {
        acc += vals[l];
    }

    const float inv = 1.0f / (float)(cnt > 0 ? cnt : 1);
    acc *= inv;

    // Write-once output: non-temporal b128 store, 256B coalesced per bag.
    v4f* op = reinterpret_cast<v4f*>(out + (size_t)bag * DD + (size_t)sub * 4);
    __builtin_nontemporal_store(acc, op);
}

extern "C" void kernel_launch(void* const* d_in, const int* in_sizes, int n_in,
                              void* d_out, int out_size, void* d_ws, size_t ws_size,
                              hipStream_t stream) {
    const int*   feats = (const int*)  d_in[0];  // int_feats [B, 1000]
    const float* Wreg  = (const float*)d_in[1];  // [25, 50001, 64]
    const float* Wcmp  = (const float*)d_in[2];  // [25, 100001, 64]
    float*       out   = (float*)      d_out;    // [B, 50, 64]

    const int B     = in_sizes[0] / (F_TOT * LL);
    const int nBags = B * F_TOT;                 // 204800 for B=4096

    // 256 threads = 8 waves = 16 bags per block.
    const int bagsPerBlock = 16;
    const int blocks = (nBags + bagsPerBlock - 1) / bagsPerBlock;

    emb_bag_kernel<<<blocks, 256, 0, stream>>>(feats, Wreg, Wcmp, out, nBags);
}